// LossNegSampling_71416716197928
// MI455X (gfx1250) — compile-verified
//
#include <hip/hip_runtime.h>
#include <hip/hip_bf16.h>

typedef __attribute__((ext_vector_type(2))) float v2f;
typedef __attribute__((ext_vector_type(8))) float v8f;

#define EMB  128
#define NLAB 64
#define NNEG 5
#define WAVES 8

__device__ __forceinline__ float logsigf(float x) {
    // log(sigmoid(x)) = min(x,0) - log1p(exp(-|x|))
    return fminf(x, 0.0f) - __logf(1.0f + __expf(-fabsf(x)));
}

__device__ __forceinline__ void wave_lds_fence() {
    __builtin_amdgcn_wave_barrier();
    asm volatile("s_wait_dscnt 0" ::: "memory");
    __builtin_amdgcn_wave_barrier();
}

__global__ void init_ws(float* ws) {
    if (threadIdx.x < 2) ws[threadIdx.x] = 0.0f;
}

__global__ void __launch_bounds__(256)
fused_kernel(const int* __restrict__ u_node, const int* __restrict__ v_node,
             const int* __restrict__ neg_nodes, const float* __restrict__ emb_u,
             const float* __restrict__ emb_com, float* __restrict__ ws, int tiles)
{
    __shared__ float comL[NLAB * EMB];      // 32 KB: emb_com resident
    __shared__ float comSq[NLAB];
    __shared__ float uSlab[WAVES][16 * EMB]; // 8 KB per wave

    const int t = threadIdx.x;
    // Stage emb_com into LDS (coalesced b128)
    for (int j = t; j < NLAB * EMB / 4; j += 256)
        *(float4*)&comL[j * 4] = *(const float4*)&emb_com[j * 4];
    __syncthreads();
    if (t < NLAB) {
        float s = 0.f;
        for (int d = 0; d < EMB; ++d) { float c = comL[t * EMB + d]; s += c * c; }
        comSq[t] = s;
    }
    __syncthreads();

    const int wave = t >> 5, lane = t & 31;
    const int half = lane >> 4, r = lane & 15;
    const int tile = blockIdx.x * WAVES + wave;
    if (tile >= tiles) return;

    const int s  = tile * 16 + r;   // sample owned by lane pair (r, r+16)
    const int d0 = half * 64;       // this lane's half of the 128-dim row

    const int iu = u_node[s];
    const int iv = v_node[s];
    const float* pu = emb_u + (size_t)iu * EMB + d0;
    const float* pv = emb_u + (size_t)iv * EMB + d0;
    const float* pn[NNEG];
    #pragma unroll
    for (int m = 0; m < NNEG; ++m)
        pn[m] = emb_u + (size_t)neg_nodes[s * NNEG + m] * EMB + d0;

    __builtin_prefetch(pu, 0, 0);
    __builtin_prefetch(pv, 0, 0);
    #pragma unroll
    for (int m = 0; m < NNEG; ++m) __builtin_prefetch(pn[m], 0, 0);

    // Single pass over gathered rows: dots + stage u into LDS for WMMA
    float usq = 0.f, pos = 0.f, nsum = 0.f;
    float* uDst = &uSlab[wave][r * EMB + d0];
    #pragma unroll
    for (int c = 0; c < 64; c += 4) {
        float4 uu = *(const float4*)(pu + c);
        *(float4*)(uDst + c) = uu;
        float4 vv = *(const float4*)(pv + c);
        usq += uu.x*uu.x + uu.y*uu.y + uu.z*uu.z + uu.w*uu.w;
        pos += uu.x*vv.x + uu.y*vv.y + uu.z*vv.z + uu.w*vv.w;
        #pragma unroll
        for (int m = 0; m < NNEG; ++m) {
            float4 nn = *(const float4*)(pn[m] + c);
            nsum += uu.x*nn.x + uu.y*nn.y + uu.z*nn.z + uu.w*nn.w;
        }
    }
    usq  += __shfl_xor(usq, 16, 32);
    pos  += __shfl_xor(pos, 16, 32);
    nsum += __shfl_xor(nsum, 16, 32);

    // negative_score = -sum(dot(u, neg_emb)); loss contribution per sample
    float ls = (half == 0) ? (logsigf(pos) + logsigf(-nsum)) : 0.0f;
    #pragma unroll
    for (int m = 16; m >= 1; m >>= 1) ls += __shfl_xor(ls, m, 32);
    if (lane == 0) atomicAdd(&ws[0], ls);

    wave_lds_fence();  // slab written cross-lane; fence before WMMA reads

    // D[m][n] = dot(u_m, com_n): 16x64 tile, K=128 via V_WMMA_F32_16X16X4_F32
    v8f a0 = {}, a1 = {}, a2 = {}, a3 = {};
    const float* uA = &uSlab[wave][r * EMB + 2 * half]; // A: M=r, K folded in halves
    const float* cB = &comL[r * EMB + 2 * half];        // B: N=r(+16t), K in halves
    #pragma unroll 4
    for (int k = 0; k < EMB; k += 4) {
        v2f a  = *(const v2f*)(uA + k);
        v2f b0 = *(const v2f*)(cB + k);
        v2f b1 = *(const v2f*)(cB + 16 * EMB + k);
        v2f b2 = *(const v2f*)(cB + 32 * EMB + k);
        v2f b3 = *(const v2f*)(cB + 48 * EMB + k);
        a0 = __builtin_amdgcn_wmma_f32_16x16x4_f32(false, a, false, b0, (short)0, a0, false, false);
        a1 = __builtin_amdgcn_wmma_f32_16x16x4_f32(false, a, false, b1, (short)0, a1, false, false);
        a2 = __builtin_amdgcn_wmma_f32_16x16x4_f32(false, a, false, b2, (short)0, a2, false, false);
        a3 = __builtin_amdgcn_wmma_f32_16x16x4_f32(false, a, false, b3, (short)0, a3, false, false);
    }

    // sq_dist[m][n] = usq[m] + comSq[n] - 2*dot ; min over n, clamp >= 0
    float q0 = comSq[r], q1 = comSq[r + 16], q2 = comSq[r + 32], q3 = comSq[r + 48];
    float minv[8];
    #pragma unroll
    for (int i = 0; i < 8; ++i) {
        float c0 = q0 - 2.f * a0[i];
        float c1 = q1 - 2.f * a1[i];
        float c2 = q2 - 2.f * a2[i];
        float c3 = q3 - 2.f * a3[i];
        minv[i] = fminf(fminf(c0, c1), fminf(c2, c3));
    }
    #pragma unroll
    for (int m = 1; m < 16; m <<= 1) {
        #pragma unroll
        for (int i = 0; i < 8; ++i)
            minv[i] = fminf(minv[i], __shfl_xor(minv[i], m, 32));
    }
    float csum = 0.f;
    #pragma unroll
    for (int i = 0; i < 8; ++i) {
        int row = half * 8 + i;              // rows 0..7 in low half, 8..15 in high
        float us = __shfl(usq, row, 32);     // usq[sample row] lives in lane 'row'
        csum += fmaxf(us + minv[i], 0.0f);
    }
    if (r == 0) atomicAdd(&ws[1], csum);
}

__global__ void finalize(const float* __restrict__ ws, float* __restrict__ out, float invB) {
    // loss = -mean(sum_all) + mean(min sq_dist) = (ws[1] - ws[0]) / B
    out[0] = (ws[1] - ws[0]) * invB;
}

extern "C" void kernel_launch(void* const* d_in, const int* in_sizes, int n_in,
                              void* d_out, int out_size, void* d_ws, size_t ws_size,
                              hipStream_t stream) {
    const int*   u_node  = (const int*)d_in[0];
    const int*   v_node  = (const int*)d_in[1];
    const int*   negs    = (const int*)d_in[2];
    const float* emb_u   = (const float*)d_in[3];
    const float* emb_com = (const float*)d_in[4];
    float* ws  = (float*)d_ws;
    float* out = (float*)d_out;

    const int batch = in_sizes[0];         // u_node is (BATCH,1)
    const int tiles = batch / 16;          // 8192 for BATCH=131072
    const int blocks = (tiles + WAVES - 1) / WAVES;

    init_ws<<<1, 256, 0, stream>>>(ws);
    fused_kernel<<<blocks, 256, 0, stream>>>(u_node, v_node, negs, emb_u, emb_com, ws, tiles);
    finalize<<<1, 1, 0, stream>>>(ws, out, 1.0f / (float)batch);
}